// DynamicConv_10934986735703
// MI455X (gfx1250) — compile-verified
//
#include <hip/hip_runtime.h>

typedef __attribute__((ext_vector_type(2))) float v2f;
typedef __attribute__((ext_vector_type(8))) float v8f;

#define CIN   256
#define COUT  48
#define HW    576           // 24*24
#define NTXT  128
#define P_W1  (CIN * COUT)          // 12288
#define P_W2  (COUT * COUT)         // 2304
#define P_O2  (P_W1 + P_W2)         // 14592
#define PDIM  (P_W1 + P_W2 + 2*COUT) // 14688
#define NTILE (PDIM / 16)           // 918

#define WMMA_F32(A, B, C) \
    __builtin_amdgcn_wmma_f32_16x16x4_f32(false, (A), false, (B), (short)0, (C), false, false)

// ---------------- Phase 1: h = relu(text @ W1 + b1) ----------------
__global__ __launch_bounds__(256) void mlp1_kernel(const float* __restrict__ text,
                                                   const float* __restrict__ W1,
                                                   const float* __restrict__ b1,
                                                   float* __restrict__ h) {
    __shared__ float ts[CIN];
    int n = blockIdx.x;
    int j = threadIdx.x;
    ts[j] = text[n * CIN + j];
    __syncthreads();
    float acc = b1[j];
#pragma unroll 4
    for (int k = 0; k < CIN; ++k) acc += ts[k] * W1[k * CIN + j];
    h[n * CIN + j] = fmaxf(acc, 0.f);
}

// ---------------- Phase 2: params = h @ W2 + b2  (128 x 14688, K=256) ----------------
__global__ __launch_bounds__(256) void mlp2_kernel(const float* __restrict__ h,
                                                   const float* __restrict__ W2,
                                                   const float* __restrict__ b2,
                                                   float* __restrict__ params) {
    __shared__ float hs[16 * 257];
    int tid = threadIdx.x;
    int mt  = blockIdx.x;                       // 8 m-tiles of 16 rows
    for (int i = tid; i < 16 * CIN; i += 256) {
        int r = i >> 8, c = i & 255;
        hs[r * 257 + c] = h[(mt * 16 + r) * CIN + c];
    }
    __syncthreads();

    int lane = tid & 31;
    int wave = tid >> 5;
    int nt   = blockIdx.y * 8 + wave;           // N-tile (16 cols) per wave
    if (nt >= NTILE) return;

    int lm   = lane & 15;
    int koff = (lane >> 4) << 1;                // lanes 16-31 handle K+2,K+3
    int j    = nt * 16 + lm;

    v8f acc = {};
#pragma unroll 8
    for (int k = 0; k < CIN; k += 4) {
        int kk = k + koff;
        v2f a, b;
        a.x = hs[lm * 257 + kk];
        a.y = hs[lm * 257 + kk + 1];
        b.x = W2[(size_t)kk * PDIM + j];
        b.y = W2[(size_t)(kk + 1) * PDIM + j];
        acc = WMMA_F32(a, b, acc);
    }
    float bj = b2[j];
    int mh = (lane >> 4) << 3;                  // upper half lanes -> rows +8
#pragma unroll
    for (int r = 0; r < 8; ++r) {
        int m = mt * 16 + r + mh;
        params[(size_t)m * PDIM + j] = acc[r] + bj;
    }
}

// ---------------- Phase 3: dynamic conv + LN + reshuffle ----------------
// grid = 8*128 blocks (one per (b,n)), 256 threads = 8 waves.
__global__ __launch_bounds__(256) void dynconv_kernel(const float* __restrict__ img,
                                                      const float* __restrict__ params,
                                                      const float* __restrict__ gamma,
                                                      const float* __restrict__ beta,
                                                      float* __restrict__ out) {
    extern __shared__ float smem[];
    float* w1s    = smem;                       // 48 x 257 (pad kills bank conflicts)
    float* w2s    = w1s + 48 * 257;             // 48 x 49
    float* bc1s   = w2s + 48 * 49;              // 48
    float* bc2s   = bc1s + 48;                  // 48
    float* gs     = bc2s + 48;                  // 48
    float* bs     = gs + 48;                    // 48
    float* xb_all = bs + 48;                    // 8 waves x 48 x 17
    float* st_all = xb_all + 8 * 48 * 17;       // 8 waves x (16 mu + 16 rsig)

    int tid = threadIdx.x;
    int n = blockIdx.x & 127;
    int b = blockIdx.x >> 7;
    const float* P = params + (size_t)n * PDIM;
    const float* I = img + (size_t)b * CIN * HW;

    for (int i = tid; i < P_W1; i += 256)
        w1s[(i >> 8) * 257 + (i & 255)] = P[i];
    for (int i = tid; i < P_W2; i += 256)
        w2s[(i / COUT) * 49 + (i % COUT)] = P[P_W1 + i];
    if (tid < COUT) {
        bc1s[tid] = P[P_O2 + tid];
        bc2s[tid] = P[P_O2 + COUT + tid];
        gs[tid]   = gamma[tid];
        bs[tid]   = beta[tid];
    }
    __syncthreads();

    int lane = tid & 31;
    int wave = tid >> 5;
    int lm   = lane & 15;
    int koff = (lane >> 4) << 1;
    int mh   = (lane >> 4) << 3;
    float* xb  = xb_all + wave * (48 * 17);
    float* mus = st_all + wave * 32;
    float* rss = mus + 16;

    int g = b >> 1, p = b & 1;                  // output reshuffle: b -> (g, p)
    float* OB = out + ((size_t)(g * NTXT + n) * 96 + p * 48) * HW;

    for (int ct = wave; ct < HW / 16; ct += 8) {
        int col0 = ct * 16;
        const float* Icol = I + col0 + lm;

        // ---- GEMM1: x1 = w1[n] (48x256) @ img_b[:, col0:col0+16]
        v8f acc0 = {}, acc1 = {}, acc2 = {};
#pragma unroll 4
        for (int k = 0; k < CIN; k += 4) {
            int kk = k + koff;
            v2f bv, a0, a1, a2;
            bv.x = Icol[(size_t)kk * HW];
            bv.y = Icol[(size_t)(kk + 1) * HW];
            a0.x = w1s[lm * 257 + kk];        a0.y = w1s[lm * 257 + kk + 1];
            a1.x = w1s[(16 + lm) * 257 + kk]; a1.y = w1s[(16 + lm) * 257 + kk + 1];
            a2.x = w1s[(32 + lm) * 257 + kk]; a2.y = w1s[(32 + lm) * 257 + kk + 1];
            acc0 = WMMA_F32(a0, bv, acc0);
            acc1 = WMMA_F32(a1, bv, acc1);
            acc2 = WMMA_F32(a2, bv, acc2);
        }
        // bias + relu -> per-wave LDS tile (48 x 16)
#pragma unroll
        for (int r = 0; r < 8; ++r) {
            int m0 = r + mh;
            xb[m0 * 17 + lm]        = fmaxf(acc0[r] + bc1s[m0], 0.f);
            xb[(m0 + 16) * 17 + lm] = fmaxf(acc1[r] + bc1s[m0 + 16], 0.f);
            xb[(m0 + 32) * 17 + lm] = fmaxf(acc2[r] + bc1s[m0 + 32], 0.f);
        }

        // ---- GEMM2: x2 = w2[n] (48x48) @ x1 (48x16)
        v8f c0 = {}, c1 = {}, c2 = {};
#pragma unroll
        for (int k = 0; k < COUT; k += 4) {
            int kk = k + koff;
            v2f bv, a0, a1, a2;
            bv.x = xb[kk * 17 + lm];
            bv.y = xb[(kk + 1) * 17 + lm];
            a0.x = w2s[lm * 49 + kk];        a0.y = w2s[lm * 49 + kk + 1];
            a1.x = w2s[(16 + lm) * 49 + kk]; a1.y = w2s[(16 + lm) * 49 + kk + 1];
            a2.x = w2s[(32 + lm) * 49 + kk]; a2.y = w2s[(32 + lm) * 49 + kk + 1];
            c0 = WMMA_F32(a0, bv, c0);
            c1 = WMMA_F32(a1, bv, c1);
            c2 = WMMA_F32(a2, bv, c2);
        }
        // bias, overwrite tile with x2
#pragma unroll
        for (int r = 0; r < 8; ++r) {
            int m0 = r + mh;
            xb[m0 * 17 + lm]        = c0[r] + bc2s[m0];
            xb[(m0 + 16) * 17 + lm] = c1[r] + bc2s[m0 + 16];
            xb[(m0 + 32) * 17 + lm] = c2[r] + bc2s[m0 + 32];
        }

        // ---- LayerNorm over 48 channels, per column (wave-private)
        if (lane < 16) {
            float s = 0.f, ss = 0.f;
#pragma unroll 8
            for (int m = 0; m < COUT; ++m) {
                float v = xb[m * 17 + lane];
                s += v; ss += v * v;
            }
            float mu  = s * (1.0f / COUT);
            float var = ss * (1.0f / COUT) - mu * mu;
            mus[lane] = mu;
            rss[lane] = rsqrtf(var + 1e-5f);
        }

        // ---- affine + coalesced store (64B contiguous per 16 cols)
        for (int i = lane; i < COUT * 16; i += 32) {
            int m = i >> 4, c = i & 15;
            float v = (xb[m * 17 + c] - mus[c]) * rss[c] * gs[m] + bs[m];
            OB[(size_t)m * HW + col0 + c] = v;
        }
    }
}

#define SMEM3_BYTES ((48 * 257 + 48 * 49 + 4 * 48 + 8 * 48 * 17 + 8 * 32) * 4)

extern "C" void kernel_launch(void* const* d_in, const int* in_sizes, int n_in,
                              void* d_out, int out_size, void* d_ws, size_t ws_size,
                              hipStream_t stream) {
    (void)in_sizes; (void)n_in; (void)out_size; (void)ws_size;
    const float* img  = (const float*)d_in[0];
    const float* text = (const float*)d_in[1];
    const float* W1   = (const float*)d_in[2];
    const float* b1   = (const float*)d_in[3];
    const float* W2   = (const float*)d_in[4];
    const float* b2   = (const float*)d_in[5];
    const float* gam  = (const float*)d_in[6];
    const float* bet  = (const float*)d_in[7];
    float* out = (float*)d_out;

    float* h      = (float*)d_ws;               // 128*256 f32
    float* params = h + NTXT * CIN;             // 128*14688 f32

    mlp1_kernel<<<NTXT, 256, 0, stream>>>(text, W1, b1, h);
    mlp2_kernel<<<dim3(8, (NTILE + 7) / 8), 256, 0, stream>>>(h, W2, b2, params);
    dynconv_kernel<<<8 * NTXT, 256, SMEM3_BYTES, stream>>>(img, params, gam, bet, out);
}